// MFNet_4423816315319
// MI455X (gfx1250) — compile-verified
//
#include <hip/hip_runtime.h>
#include <hip/hip_bf16.h>

// ---------------------------------------------------------------------------
// MFConv x2 on gfx1250: degree-bucketed GEMM via V_WMMA_F32_16X16X4_F32.
// Pipeline:
//   1) zero workspace regions
//   2) edge scatter-add:  h[dst] += x[src], deg[dst]++
//   3) bucket nodes by clamped degree (11 buckets)
//   4) per-bucket WMMA GEMM: h1 = relu([h|x] @ [W1|Wr1] + b1)   (K=256 -> 32)
//   5) edge scatter-add:  hagg2[dst] += h1[src]
//   6) per-bucket WMMA GEMM: out = [hagg2|h1] @ [W2|Wr2] + b2   (K=64 -> 64)
// ---------------------------------------------------------------------------

typedef float v2f __attribute__((ext_vector_type(2)));
typedef float v8f __attribute__((ext_vector_type(8)));

#define MAXDEG 10
#define NBUCK  11

// ---------------------------------------------------------------- zero utils
__global__ void zero_f32(float* __restrict__ p, size_t n) {
    size_t i = (size_t)blockIdx.x * blockDim.x + threadIdx.x;
    size_t stride = (size_t)gridDim.x * blockDim.x;
    for (; i < n; i += stride) p[i] = 0.0f;
}

__global__ void zero_i32(int* __restrict__ p, size_t n) {
    size_t i = (size_t)blockIdx.x * blockDim.x + threadIdx.x;
    size_t stride = (size_t)gridDim.x * blockDim.x;
    for (; i < n; i += stride) p[i] = 0;
}

// ------------------------------------------------- layer-1 aggregation + deg
// one wave (32 lanes) per edge: 128 features scatter-added with f32 atomics
__global__ __launch_bounds__(256)
void mf_deg_agg(const float* __restrict__ x, const long long* __restrict__ ei,
                float* __restrict__ h, int* __restrict__ deg, int E) {
    int e = blockIdx.x * 8 + (threadIdx.x >> 5);
    int lane = threadIdx.x & 31;
    if (e >= E) return;
    int s = (int)ei[e];
    int d = (int)ei[(size_t)E + e];
    if (lane == 0) atomicAdd(&deg[d], 1);
    const float* xs = x + (size_t)s * 128;
    float*       hd = h + (size_t)d * 128;
#pragma unroll
    for (int f = lane; f < 128; f += 32) atomicAdd(&hd[f], xs[f]);
}

// --------------------------------------------------------- bucket node lists
__global__ __launch_bounds__(256)
void mf_bucket(const int* __restrict__ deg, int* __restrict__ counts,
               int* __restrict__ lists, int N) {
    int n = blockIdx.x * 256 + threadIdx.x;
    if (n >= N) return;
    int d = deg[n];
    if (d > MAXDEG) d = MAXDEG;
    int pos = atomicAdd(&counts[d], 1);
    lists[(size_t)d * N + pos] = n;
}

// ------------------------------------------------------------ layer-1 WMMA
// out[16,32] = [h|x][16,256] @ [W1[d]|Wr1[d]][256,32] + b1[d]; relu -> h1
__global__ __launch_bounds__(32)
void mf_layer1_wmma(const float* __restrict__ h, const float* __restrict__ x,
                    const float* __restrict__ W1, const float* __restrict__ b1,
                    const float* __restrict__ Wr1,
                    const int* __restrict__ counts, const int* __restrict__ lists,
                    float* __restrict__ h1, int N) {
    const int d = blockIdx.y;
    const int cnt = counts[d];
    const int t = blockIdx.x;
    if (t * 16 >= cnt) return;
    const int rows = min(16, cnt - t * 16);

    __shared__ int   nid[16];
    __shared__ float As[16][260];   // 16 x 256, padded stride
    __shared__ float Bs[256][33];   // 256 x 32, padded stride

    const int lane = threadIdx.x;
    if (lane < 16) {
        int idx = t * 16 + lane;
        nid[lane] = lists[(size_t)d * N + ((idx < cnt) ? idx : t * 16)];
    }
    __syncthreads();

    // stage A tile: row r = concat(h[n_r], x[n_r])
    for (int i = lane; i < 16 * 256; i += 32) {
        int r = i >> 8, f = i & 255;
        int n = nid[r];
        As[r][f] = (f < 128) ? h[(size_t)n * 128 + f]
                             : x[(size_t)n * 128 + (f - 128)];
    }
    // stage B: concat(W1[d], Wr1[d]) along K, 256x32
    const float* Wd  = W1  + (size_t)d * 128 * 32;
    const float* Wrd = Wr1 + (size_t)d * 128 * 32;
    for (int i = lane; i < 256 * 32; i += 32) {
        int kk = i >> 5, c = i & 31;
        Bs[kk][c] = (kk < 128) ? Wd[kk * 32 + c] : Wrd[(kk - 128) * 32 + c];
    }
    __syncthreads();

    const int hi = lane >> 4, lo = lane & 15;
    v8f acc0 = {}; v8f acc1 = {};
#pragma unroll 4
    for (int k = 0; k < 256; k += 4) {
        int kb = k + 2 * hi;
        v2f a;  a.x  = As[lo][kb];      a.y  = As[lo][kb + 1];
        v2f b0; b0.x = Bs[kb][lo];      b0.y = Bs[kb + 1][lo];
        v2f bq; bq.x = Bs[kb][16 + lo]; bq.y = Bs[kb + 1][16 + lo];
        acc0 = __builtin_amdgcn_wmma_f32_16x16x4_f32(false, a, false, b0,
                                                     (short)0, acc0, false, false);
        acc1 = __builtin_amdgcn_wmma_f32_16x16x4_f32(false, a, false, bq,
                                                     (short)0, acc1, false, false);
    }

    const float bias0 = b1[d * 32 + lo];
    const float bias1 = b1[d * 32 + 16 + lo];
#pragma unroll
    for (int v = 0; v < 8; ++v) {
        int row = v + 8 * hi;                 // C/D layout: M = v + 8*(lane/16)
        if (row < rows) {
            int n = nid[row];
            float o0 = acc0[v] + bias0; o0 = o0 > 0.0f ? o0 : 0.0f;
            float o1 = acc1[v] + bias1; o1 = o1 > 0.0f ? o1 : 0.0f;
            h1[(size_t)n * 32 + lo]      = o0;
            h1[(size_t)n * 32 + 16 + lo] = o1;
        }
    }
}

// ------------------------------------------------- layer-2 aggregation
__global__ __launch_bounds__(256)
void mf_agg2(const float* __restrict__ h1, const long long* __restrict__ ei,
             float* __restrict__ hagg2, int E) {
    int e = blockIdx.x * 8 + (threadIdx.x >> 5);
    int lane = threadIdx.x & 31;
    if (e >= E) return;
    int s = (int)ei[e];
    int d = (int)ei[(size_t)E + e];
    atomicAdd(&hagg2[(size_t)d * 32 + lane], h1[(size_t)s * 32 + lane]);
}

// ------------------------------------------------------------ layer-2 WMMA
// out[16,64] = [hagg2|h1][16,64] @ [W2[d]|Wr2[d]][64,64] + b2[d]
__global__ __launch_bounds__(32)
void mf_layer2_wmma(const float* __restrict__ hagg2, const float* __restrict__ h1,
                    const float* __restrict__ W2, const float* __restrict__ b2,
                    const float* __restrict__ Wr2,
                    const int* __restrict__ counts, const int* __restrict__ lists,
                    float* __restrict__ out, int N) {
    const int d = blockIdx.y;
    const int cnt = counts[d];
    const int t = blockIdx.x;
    if (t * 16 >= cnt) return;
    const int rows = min(16, cnt - t * 16);

    __shared__ int   nid[16];
    __shared__ float As[16][68];    // 16 x 64
    __shared__ float Bs[64][65];    // 64 x 64

    const int lane = threadIdx.x;
    if (lane < 16) {
        int idx = t * 16 + lane;
        nid[lane] = lists[(size_t)d * N + ((idx < cnt) ? idx : t * 16)];
    }
    __syncthreads();

    for (int i = lane; i < 16 * 64; i += 32) {
        int r = i >> 6, f = i & 63;
        int n = nid[r];
        As[r][f] = (f < 32) ? hagg2[(size_t)n * 32 + f]
                            : h1[(size_t)n * 32 + (f - 32)];
    }
    const float* Wd  = W2  + (size_t)d * 32 * 64;
    const float* Wrd = Wr2 + (size_t)d * 32 * 64;
    for (int i = lane; i < 64 * 64; i += 32) {
        int kk = i >> 6, c = i & 63;
        Bs[kk][c] = (kk < 32) ? Wd[kk * 64 + c] : Wrd[(kk - 32) * 64 + c];
    }
    __syncthreads();

    const int hi = lane >> 4, lo = lane & 15;
    v8f acc[4] = {{}, {}, {}, {}};
#pragma unroll 4
    for (int k = 0; k < 64; k += 4) {
        int kb = k + 2 * hi;
        v2f a; a.x = As[lo][kb]; a.y = As[lo][kb + 1];
#pragma unroll
        for (int ct = 0; ct < 4; ++ct) {
            v2f b; b.x = Bs[kb][ct * 16 + lo]; b.y = Bs[kb + 1][ct * 16 + lo];
            acc[ct] = __builtin_amdgcn_wmma_f32_16x16x4_f32(false, a, false, b,
                                                            (short)0, acc[ct],
                                                            false, false);
        }
    }

#pragma unroll
    for (int ct = 0; ct < 4; ++ct) {
        float bias = b2[d * 64 + ct * 16 + lo];
#pragma unroll
        for (int v = 0; v < 8; ++v) {
            int row = v + 8 * hi;
            if (row < rows) {
                int n = nid[row];
                out[(size_t)n * 64 + ct * 16 + lo] = acc[ct][v] + bias;
            }
        }
    }
}

// ---------------------------------------------------------------------------
extern "C" void kernel_launch(void* const* d_in, const int* in_sizes, int n_in,
                              void* d_out, int out_size, void* d_ws, size_t ws_size,
                              hipStream_t stream) {
    const float*     x   = (const float*)d_in[0];
    const long long* ei  = (const long long*)d_in[1];   // int64 edge_index [2,E]
    const float*     W1  = (const float*)d_in[2];
    const float*     b1  = (const float*)d_in[3];
    const float*     Wr1 = (const float*)d_in[4];
    const float*     W2  = (const float*)d_in[5];
    const float*     b2  = (const float*)d_in[6];
    const float*     Wr2 = (const float*)d_in[7];
    float* out = (float*)d_out;

    const int N = in_sizes[0] / 128;
    const int E = in_sizes[1] / 2;

    // workspace layout (floats then ints)
    float* h     = (float*)d_ws;                  // N*128
    float* hagg2 = h + (size_t)N * 128;           // N*32
    float* h1    = hagg2 + (size_t)N * 32;        // N*32
    int*   deg    = (int*)(h1 + (size_t)N * 32);  // N
    int*   counts = deg + N;                      // 16 (11 used)
    int*   lists  = counts + 16;                  // 11*N

    // 1) zero h + hagg2 (contiguous 160*N floats) and deg + counts
    {
        size_t nf = (size_t)N * 160;
        int blocks = (int)((nf + 256 * 8 - 1) / (256 * 8));
        zero_f32<<<blocks, 256, 0, stream>>>(h, nf);
        size_t ni = (size_t)N + 16;
        zero_i32<<<(int)((ni + 2047) / 2048), 256, 0, stream>>>(deg, ni);
    }

    // 2) layer-1 aggregation + degree
    mf_deg_agg<<<(E + 7) / 8, 256, 0, stream>>>(x, ei, h, deg, E);

    // 3) bucket nodes by degree
    mf_bucket<<<(N + 255) / 256, 256, 0, stream>>>(deg, counts, lists, N);

    // 4) layer-1 GEMM (WMMA) + relu
    const int tiles = (N + 15) / 16;
    mf_layer1_wmma<<<dim3(tiles, NBUCK), 32, 0, stream>>>(h, x, W1, b1, Wr1,
                                                          counts, lists, h1, N);

    // 5) layer-2 aggregation
    mf_agg2<<<(E + 7) / 8, 256, 0, stream>>>(h1, ei, hagg2, E);

    // 6) layer-2 GEMM (WMMA) -> out
    mf_layer2_wmma<<<dim3(tiles, NBUCK), 32, 0, stream>>>(hagg2, h1, W2, b2, Wr2,
                                                          counts, lists, out, N);
}